// expressimg_21655225107066
// MI455X (gfx1250) — compile-verified
//
#include <hip/hip_runtime.h>

typedef __attribute__((ext_vector_type(2))) float v2f;
typedef __attribute__((ext_vector_type(8))) float v8f;

#define HH 1024
#define WW 1024
#define NCH 16
#define HWSZ (HH * WW)
#define WAVES 4   // blocks (8x8 tiles) per workgroup, one per wave

__global__ void init_ws(unsigned* ws) { ws[0] = 0u; }

// Pass 1: global max of |x[w] - x[w-1]| (row-wise left pad with 0)
__global__ __launch_bounds__(256) void absmax_kernel(const float* __restrict__ x,
                                                     unsigned* __restrict__ ws) {
  const int total = NCH * HWSZ;
  float m = 0.0f;
  for (int i = blockIdx.x * blockDim.x + threadIdx.x; i < total;
       i += gridDim.x * blockDim.x) {
    float xi = x[i];
    float xl = ((i & (WW - 1)) != 0) ? x[i - 1] : 0.0f;
    m = fmaxf(m, fabsf(xi - xl));
  }
#pragma unroll
  for (int off = 16; off > 0; off >>= 1)
    m = fmaxf(m, __shfl_xor(m, off, 32));
  if ((threadIdx.x & 31) == 0)
    atomicMax(ws, __float_as_uint(m));  // non-negative floats: uint order == float order
}

// Pass 2: one wave per 8x8 block; quantize diffs, LS-projection via f32 WMMA, select, store.
__global__ __launch_bounds__(32 * WAVES) void block_kernel(const float* __restrict__ x,
                                                           const unsigned* __restrict__ ws,
                                                           float* __restrict__ out) {
  __shared__ float raw_s[WAVES][NCH * 8 * 9];  // 16ch x 8rows x 9cols (col0 = left neighbor)
  __shared__ float dq_s[WAVES][64 * 17];       // d[k][n], pitch 17 (bank-conflict free)
  __shared__ float p_s[WAVES][48];             // p (3x16)

  const int wave = threadIdx.x >> 5;
  const int lane = threadIdx.x & 31;
  const int bid = blockIdx.x * WAVES + wave;  // 0..16383
  const int wb = bid & 127;
  const int hb = bid >> 7;
  const int row0 = hb * 8;
  const int col0 = wb * 8 - 1;

  float* raw = raw_s[wave];
  float* dq  = dq_s[wave];
  float* pm  = p_s[wave];

  // ---- load raw tile (with left halo column) ----
  for (int idx = lane; idx < NCH * 8 * 9; idx += 32) {
    int n = idx / 72;
    int rem = idx - n * 72;
    int dy = rem / 9;
    int c = rem - dy * 9;
    int w = col0 + c;
    raw[idx] = (w >= 0) ? x[n * HWSZ + (row0 + dy) * WW + w] : 0.0f;
  }
  __syncthreads();

  const float in_max = __uint_as_float(ws[0]);
  const float lsb = exp2f(rintf(log2f(in_max * (1.0f / 128.0f))) + 1.0f);
  const float inv_lsb = 1.0f / lsb;

  // ---- quantized horizontal differences -> dq[k*17 + n] ----
  for (int idx = lane; idx < NCH * 64; idx += 32) {
    int n = idx >> 6;
    int k = idx & 63;
    int dy = k >> 3;
    int dx = k & 7;
    float xr = raw[n * 72 + dy * 9 + dx + 1];
    float xl = raw[n * 72 + dy * 9 + dx];
    dq[k * 17 + n] = rintf((xr - xl) * inv_lsb) * lsb;
  }
  __syncthreads();

  // ---- stats over a1 = dq[:,0], a2 = dq[:,1] (wave butterfly reduction) ----
  float s1 = 0, s2 = 0, s11 = 0, s22 = 0, s12 = 0;
  float mn1 = 3.4e38f, mx1 = -3.4e38f, mn2 = 3.4e38f, mx2 = -3.4e38f;
#pragma unroll
  for (int t = 0; t < 2; ++t) {
    int k = lane + 32 * t;
    float a1 = dq[k * 17 + 0], a2 = dq[k * 17 + 1];
    s1 += a1; s2 += a2; s11 += a1 * a1; s22 += a2 * a2; s12 += a1 * a2;
    mn1 = fminf(mn1, a1); mx1 = fmaxf(mx1, a1);
    mn2 = fminf(mn2, a2); mx2 = fmaxf(mx2, a2);
  }
#pragma unroll
  for (int off = 16; off > 0; off >>= 1) {
    s1 += __shfl_xor(s1, off, 32);
    s2 += __shfl_xor(s2, off, 32);
    s11 += __shfl_xor(s11, off, 32);
    s22 += __shfl_xor(s22, off, 32);
    s12 += __shfl_xor(s12, off, 32);
    mn1 = fminf(mn1, __shfl_xor(mn1, off, 32));
    mx1 = fmaxf(mx1, __shfl_xor(mx1, off, 32));
    mn2 = fminf(mn2, __shfl_xor(mn2, off, 32));
    mx2 = fmaxf(mx2, __shfl_xor(mx2, off, 32));
  }
  const bool flat = ((mx1 - mn1) < 1e-6f) && ((mx2 - mn2) < 1e-6f);

  // ---- symmetric 3x3 (A A^T) inverse; det==0 -> identity ----
  const float s33 = 64.0f;
  float det = s11 * (s22 * s33 - s2 * s2) - s12 * (s12 * s33 - s2 * s1) +
              s1 * (s12 * s2 - s22 * s1);
  float i00, i01, i02, i11, i12, i22;
  if (det == 0.0f) {
    i00 = 1; i11 = 1; i22 = 1; i01 = 0; i02 = 0; i12 = 0;
  } else {
    float rdet = 1.0f / det;
    i00 = (s22 * s33 - s2 * s2) * rdet;
    i01 = (s1 * s2 - s12 * s33) * rdet;
    i02 = (s12 * s2 - s22 * s1) * rdet;
    i11 = (s11 * s33 - s1 * s1) * rdet;
    i12 = (s1 * s12 - s11 * s2) * rdet;
    i22 = (s11 * s22 - s12 * s12) * rdet;
  }

  // base[m][k] = c0*a1[k] + c1*a2[k] + c2 for m<3, else 0 (pad rows 3..15)
  const int m = lane & 15;
  const int half = lane >> 4;
  float c0 = (m == 0) ? i00 : (m == 1) ? i01 : (m == 2) ? i02 : 0.0f;
  float c1 = (m == 0) ? i01 : (m == 1) ? i11 : (m == 2) ? i12 : 0.0f;
  float c2 = (m == 0) ? i02 : (m == 1) ? i12 : (m == 2) ? i22 : 0.0f;

  // ---- stage 1: p(16x16, rows 0..2 valid) = base(16x64) @ d(64x16) ----
  // 16 chained V_WMMA_F32_16X16X4_F32 steps over K.
  // A layout (16x4 f32): VGPR0 = K0 (lanes 0-15) / K2 (lanes 16-31), VGPR1 = K1 / K3.
  v8f pacc = {};
#pragma unroll
  for (int s = 0; s < 16; ++s) {
    int k0 = 4 * s + 2 * half;
    float a1k0 = dq[k0 * 17 + 0], a2k0 = dq[k0 * 17 + 1];
    float a1k1 = dq[(k0 + 1) * 17 + 0], a2k1 = dq[(k0 + 1) * 17 + 1];
    v2f A, B;
    A.x = c0 * a1k0 + c1 * a2k0 + c2;
    A.y = c0 * a1k1 + c1 * a2k1 + c2;
    B.x = dq[k0 * 17 + m];
    B.y = dq[(k0 + 1) * 17 + m];
    pacc = __builtin_amdgcn_wmma_f32_16x16x4_f32(false, A, false, B, (short)0,
                                                 pacc, false, false);
  }

  // p rows 0..2 live in C/D VGPRs 0..2 of lanes 0..15 -> stage via LDS
  if (lane < 16) {
    pm[0 * 16 + lane] = pacc[0];
    pm[1 * 16 + lane] = pacc[1];
    pm[2 * 16 + lane] = pacc[2];
  }
  __syncthreads();

  // ---- stage 2: r(64x16) = A^T(64x4, col3 zero) @ p(4x16, row3 zero) ----
  v2f Bp;
  Bp.x = (half == 0) ? pm[0 * 16 + m] : pm[2 * 16 + m];
  Bp.y = (half == 0) ? pm[1 * 16 + m] : 0.0f;

  v8f r[4];
#pragma unroll
  for (int t = 0; t < 4; ++t) {
    int k = 16 * t + m;
    v2f At;
    At.x = (half == 0) ? dq[k * 17 + 0] : 1.0f;  // K0=a1 | K2=ones
    At.y = (half == 0) ? dq[k * 17 + 1] : 0.0f;  // K1=a2 | K3=pad
    v8f z = {};
    r[t] = __builtin_amdgcn_wmma_f32_16x16x4_f32(false, At, false, Bp, (short)0,
                                                 z, false, false);
  }

  // ---- quantize r, per-channel loss ----
  float loss = 0.0f;
#pragma unroll
  for (int t = 0; t < 4; ++t) {
#pragma unroll
    for (int j = 0; j < 8; ++j) {
      int k = 16 * t + j + 8 * half;  // C/D layout: VGPR j = row j (lanes<16) / j+8
      float r1 = rintf(r[t][j] * inv_lsb) * lsb;
      float dv = dq[k * 17 + m];
      float e = dv - r1;
      loss += e * e;
      r[t][j] = r1;
    }
  }
  loss += __shfl_xor(loss, 16, 32);  // lanes n and n+16 both get channel-n loss
  const bool use_r1 = (!flat) && (loss <= 10.0f);

  // write rr back into dq (each lane owns disjoint (k,m) slots)
#pragma unroll
  for (int t = 0; t < 4; ++t) {
#pragma unroll
    for (int j = 0; j < 8; ++j) {
      int k = 16 * t + j + 8 * half;
      float dv = dq[k * 17 + m];
      dq[k * 17 + m] = use_r1 ? r[t][j] : dv;
    }
  }
  __syncthreads();

  // ---- final: out = rr + x_left, back to planar (N,H,W) layout ----
  for (int idx = lane; idx < NCH * 64; idx += 32) {
    int n = idx >> 6;
    int k = idx & 63;
    int dy = k >> 3;
    int dx = k & 7;
    float left = raw[n * 72 + dy * 9 + dx];  // x at w-1
    out[n * HWSZ + (row0 + dy) * WW + (wb * 8 + dx)] = dq[k * 17 + n] + left;
  }
}

extern "C" void kernel_launch(void* const* d_in, const int* in_sizes, int n_in,
                              void* d_out, int out_size, void* d_ws, size_t ws_size,
                              hipStream_t stream) {
  const float* x = (const float*)d_in[0];
  float* out = (float*)d_out;
  unsigned* ws = (unsigned*)d_ws;

  hipLaunchKernelGGL(init_ws, dim3(1), dim3(1), 0, stream, ws);
  hipLaunchKernelGGL(absmax_kernel, dim3(2048), dim3(256), 0, stream, x, ws);
  hipLaunchKernelGGL(block_kernel, dim3(16384 / WAVES), dim3(32 * WAVES), 0, stream,
                     x, ws, out);
}